// SynonymAwareCrossEntropyLoss_3066606649478
// MI455X (gfx1250) — compile-verified
//
#include <hip/hip_runtime.h>
#include <math.h>

// ---------------- synonym tables (compile-time constants from the module) ----------------
#define NKEYS 11
#define MAXSYN 5
__constant__ int cKeys[NKEYS] = {3363,1400,1029,7090,1877,6045,10073,5882,1097,5719,1323};
__constant__ int cIds[NKEYS][MAXSYN] = {
  {3763,10889,826,2081,3376},   // 3363
  {645,407,10352,1239,8005},    // 1400
  {3334,0,0,0,0},               // 1029
  {13398,0,0,0,0},              // 7090
  {7754,0,0,0,0},               // 1877
  {1239,4844,0,0,0},            // 6045
  {1790,1449,0,0,0},            // 10073
  {890,0,0,0,0},                // 5882
  {1097,0,0,0,0},               // 1097
  {5719,0,0,0,0},               // 5719
  {1323,0,0,0,0}};              // 1323
__constant__ int cCnt[NKEYS] = {5,5,1,1,1,2,2,1,1,1,1};

// ---------------- CDNA5 async global->LDS copy path ----------------
// Non-temporal: the logit stream (412 MB) is single-use and bigger than the
// 192 MB L2, so tell WGP$/L2 not to retain the lines.
__device__ __forceinline__ void async_copy_b128(const float4* gsrc, const float4* ldst) {
  // Low 32 bits of a generic pointer to __shared__ are the LDS byte offset
  // (flat->LDS mapping truncates to addr[31:0], ISA 10.2).
  unsigned loff = (unsigned)(uintptr_t)ldst;
  asm volatile("global_load_async_to_lds_b128 %0, %1, off th:TH_LOAD_NT"
               :: "v"(loff), "v"(gsrc) : "memory");
}
__device__ __forceinline__ void wait_async_le3() {
  asm volatile("s_wait_asynccnt 0x3" ::: "memory");
}
__device__ __forceinline__ void wait_async_le2() {
  asm volatile("s_wait_asynccnt 0x2" ::: "memory");
}
__device__ __forceinline__ void wait_async_le1() {
  asm volatile("s_wait_asynccnt 0x1" ::: "memory");
}
__device__ __forceinline__ void wait_async_le0() {
  asm volatile("s_wait_asynccnt 0x0" ::: "memory");
}

__device__ __forceinline__ float e2(float x) { return __builtin_amdgcn_exp2f(x); }

#define L2E 1.4426950408889634f
#define LN2 0.6931471805599453f
#define BLK 256
#define NBUF 4   // async pipeline depth (4 x 4 KB chunks in flight per block)

// Online-softmax pair combine: (m,s) <- (m,s) + (om,os)
__device__ __forceinline__ void combine(float& m, float& s, float om, float os) {
  float nm = fmaxf(m, om);
  s = s * e2((m - nm) * L2E) + os * e2((om - nm) * L2E);
  m = nm;
}

__global__ __launch_bounds__(BLK)
void SynCE_row_kernel(const float* __restrict__ logits,
                      const int* __restrict__ labels,
                      int V, float* __restrict__ tot, float* __restrict__ val) {
  const int row = blockIdx.x;
  const int tid = threadIdx.x;
  const float* rp = logits + (size_t)row * (size_t)V;
  const float4* rp4 = (const float4*)rp;

  __shared__ float4 buf[NBUF][BLK];   // 16 KB circular buffer

  const int nchunks = V >> 10;                 // 1024 floats (BLK * float4) per chunk
  const int tail_base = nchunks << 10;
  const int tail = V - tail_base;

  float m = -INFINITY;
  float s = 0.0f;

  // ---- prologue: fill the pipeline (up to NBUF chunks in flight) ----
  if (nchunks > 0) async_copy_b128(rp4 + 0 * BLK + tid, &buf[0][tid]);
  if (nchunks > 1) async_copy_b128(rp4 + 1 * BLK + tid, &buf[1][tid]);
  if (nchunks > 2) async_copy_b128(rp4 + 2 * BLK + tid, &buf[2][tid]);
  if (nchunks > 3) async_copy_b128(rp4 + 3 * BLK + tid, &buf[3][tid]);

  for (int c = 0; c < nchunks; ++c) {
    // Async loads retire in order: chunk c is in LDS once the number of
    // still-outstanding (younger) loads drops to min(nchunks-1-c, 3).
    const int rem = nchunks - 1 - c;
    if (rem >= 3)      wait_async_le3();
    else if (rem == 2) wait_async_le2();
    else if (rem == 1) wait_async_le1();
    else               wait_async_le0();

    float4 x = buf[c & (NBUF - 1)][tid];   // ds_load_b128
    float cm = fmaxf(fmaxf(x.x, x.y), fmaxf(x.z, x.w));
    float nm = fmaxf(m, cm);
    float sc = e2((m - nm) * L2E);
    s = s * sc
      + e2((x.x - nm) * L2E) + e2((x.y - nm) * L2E)
      + e2((x.z - nm) * L2E) + e2((x.w - nm) * L2E);
    m = nm;

    // Refill the buffer we just consumed (safe: the ds_load above is pinned
    // before this issue by the asm "memory" clobbers, and completes long
    // before the global round trip returns).
    if (c + NBUF < nchunks)
      async_copy_b128(rp4 + (size_t)(c + NBUF) * BLK + tid, &buf[c & (NBUF - 1)][tid]);
  }

  // tail (V % 1024 elements), plain global loads
  if (tid < tail) {
    float x = rp[tail_base + tid];
    float nm = fmaxf(m, x);
    s = s * e2((m - nm) * L2E) + e2((x - nm) * L2E);
    m = nm;
  }

  // ---- intra-wave reduction (wave32) ----
  #pragma unroll
  for (int off = 16; off > 0; off >>= 1) {
    float om = __shfl_xor(m, off, 32);
    float os = __shfl_xor(s, off, 32);
    combine(m, s, om, os);
  }

  // ---- cross-wave reduction (8 waves) ----
  __shared__ float sm[BLK / 32], ss[BLK / 32];
  const int wid = tid >> 5, lane = tid & 31;
  if (lane == 0) { sm[wid] = m; ss[wid] = s; }
  __syncthreads();
  if (wid == 0) {
    m = (lane < (BLK / 32)) ? sm[lane] : -INFINITY;
    s = (lane < (BLK / 32)) ? ss[lane] : 0.0f;
    #pragma unroll
    for (int off = (BLK / 64); off > 0; off >>= 1) {
      float om = __shfl_xor(m, off, 32);
      float os = __shfl_xor(s, off, 32);
      combine(m, s, om, os);
    }
    if (lane == 0) {
      const float lnS = __builtin_amdgcn_logf(s) * LN2;  // ln(sum exp(x - m))
      const int lbl = labels[row];
      float total = 0.0f, valid = 0.0f;
      if (lbl != -100) {
        valid = 1.0f;
        const float base = -(rp[lbl] - m - lnS);
        int kidx = -1;
        #pragma unroll
        for (int k = 0; k < NKEYS; ++k)
          if (lbl == cKeys[k]) kidx = k;
        if (kidx >= 0) {
          const int cnt = cCnt[kidx];
          float synsum = 0.0f;
          for (int j = 0; j < cnt; ++j)
            synsum += -(rp[cIds[kidx][j]] - m - lnS);
          total = 0.8f * base + synsum * (0.2f / (float)cnt);
        } else {
          total = base;
        }
      }
      tot[row] = total * valid;
      val[row] = valid;
    }
  }
}

__global__ __launch_bounds__(BLK)
void SynCE_finalize_kernel(const float* __restrict__ tot,
                           const float* __restrict__ val,
                           int n, float* __restrict__ out) {
  double ts = 0.0, vs = 0.0;
  for (int i = threadIdx.x; i < n; i += BLK) { ts += (double)tot[i]; vs += (double)val[i]; }
  #pragma unroll
  for (int off = 16; off > 0; off >>= 1) {
    ts += __shfl_xor(ts, off, 32);
    vs += __shfl_xor(vs, off, 32);
  }
  __shared__ double st[BLK / 32], sv[BLK / 32];
  const int wid = threadIdx.x >> 5, lane = threadIdx.x & 31;
  if (lane == 0) { st[wid] = ts; sv[wid] = vs; }
  __syncthreads();
  if (threadIdx.x == 0) {
    double T = 0.0, Vv = 0.0;
    #pragma unroll
    for (int w = 0; w < BLK / 32; ++w) { T += st[w]; Vv += sv[w]; }
    out[0] = (float)(T / Vv);
  }
}

extern "C" void kernel_launch(void* const* d_in, const int* in_sizes, int n_in,
                              void* d_out, int out_size, void* d_ws, size_t ws_size,
                              hipStream_t stream) {
  const float* logits = (const float*)d_in[0];
  const int*   labels = (const int*)d_in[1];
  const int N = in_sizes[1];               // 2048 tokens
  const int V = in_sizes[0] / N;           // 50257
  float* tot = (float*)d_ws;               // [N]
  float* val = tot + N;                    // [N]
  float* out = (float*)d_out;              // scalar

  SynCE_row_kernel<<<N, BLK, 0, stream>>>(logits, labels, V, tot, val);
  SynCE_finalize_kernel<<<1, BLK, 0, stream>>>(tot, val, N, out);
}